// MoiraiGating_14516989460786
// MI455X (gfx1250) — compile-verified
//
#include <hip/hip_runtime.h>

// CDNA5 / gfx1250, wave32. f32 WMMA (V_WMMA_F32_16X16X4_F32) + TDM async tile DMA.
typedef float v2f __attribute__((ext_vector_type(2)));
typedef float v8f __attribute__((ext_vector_type(8)));
typedef unsigned int u32x4 __attribute__((ext_vector_type(4)));
typedef int          i32x4 __attribute__((ext_vector_type(4)));
typedef int          i32x8 __attribute__((ext_vector_type(8)));

#define DIM        2048   // input_dim (K)
#define NEXP       64     // experts (N)
#define KC         128    // K-chunk staged in LDS per stage
#define NSTAGES    (DIM / KC)
#define XSTRIDE    132    // padded LDS row stride (floats) == 128 + 4 (TDM pad 4 DW / 128 DW)
#define LGSTRIDE   65     // padded logit row stride -> conflict-free top-k scan
#define WG_TOKENS  64     // tokens per workgroup (4 waves x 16 tokens)
#define NTHREADS   128

#ifndef __has_builtin
#define __has_builtin(x) 0
#endif

#if __has_builtin(__builtin_amdgcn_tensor_load_to_lds) && \
    __has_builtin(__builtin_amdgcn_s_wait_tensorcnt)
#define USE_TDM 1
#if __has_include(<hip/amd_detail/amd_gfx1250_TDM.h>)
#define TDM_SIX_ARGS 1   // amdgpu-toolchain (clang-23 / therock-10.0) lane
#endif
#endif

#ifdef USE_TDM
// Issue one TDM 2D-tile load: tile0 x tile1 elements (4B each) from a row-major
// tensor with row stride `stride0` elements, into LDS at byte offset `lds_off`,
// inserting 4 DWORDs of padding after every 128 DWORDs (-> XSTRIDE=132 rows).
__device__ __forceinline__ void tdm_load_tile_2d(const void* gptr, unsigned lds_off,
                                                 unsigned td0, unsigned td1,
                                                 unsigned tile0, unsigned tile1,
                                                 unsigned long long stride0)
{
    unsigned long long ga = (unsigned long long)(uintptr_t)gptr;
    u32x4 g0;
    g0[0] = 1u;                                            // count=1, user descriptor
    g0[1] = lds_off;                                       // lds_addr (bytes)
    g0[2] = (unsigned)(ga & 0xFFFFFFFFu);                  // global_addr[31:0]
    g0[3] = (unsigned)((ga >> 32) & 0x01FFFFFFu)           // global_addr[56:32]
          | (2u << 30);                                    // type=2 ("image")
    i32x8 g1;
    g1[0] = (int)((2u << 16)        // data_size = 4 bytes
                | (1u << 20)        // pad_enable
                | (6u << 22)        // pad_interval: 128 DWORDs
                | (3u << 25));      // pad_amount : 4 DWORDs
    g1[1] = (int)((td0 & 0xFFFFu) << 16);                  // atomic_barrier_addr=0 | td0.lo
    g1[2] = (int)((td0 >> 16) | ((td1 & 0xFFFFu) << 16));  // td0.hi | td1.lo
    g1[3] = (int)((td1 >> 16) | (tile0 << 16));            // td1.hi | tile_dim0
    g1[4] = (int)(tile1 & 0xFFFFu);                        // tile_dim1 | tile_dim2=0
    g1[5] = (int)(stride0 & 0xFFFFFFFFull);                // tensor_dim0_stride[31:0]
    g1[6] = (int)((stride0 >> 32) & 0xFFFFull);            // [47:32] | dim1_stride.lo=0
    g1[7] = 0;
    i32x4 gz4 = {0, 0, 0, 0};                              // groups 2/3: unused dims
#ifdef TDM_SIX_ARGS
    i32x8 gz8 = {0, 0, 0, 0, 0, 0, 0, 0};
    __builtin_amdgcn_tensor_load_to_lds(g0, g1, gz4, gz4, gz8, 0);
#else
    __builtin_amdgcn_tensor_load_to_lds(g0, g1, gz4, gz4, 0);
#endif
}
#endif // USE_TDM

__global__ __launch_bounds__(NTHREADS)
void moirai_gating_kernel(const float* __restrict__ x,
                          const float* __restrict__ W,
                          const float* __restrict__ bias,
                          float* __restrict__ out_probs,
                          int*   __restrict__ out_idx)
{
    __shared__ float xs[2][WG_TOKENS * XSTRIDE];   // double-buffered 64x128 x-tiles
    __shared__ float ws[2][NEXP     * XSTRIDE];    // double-buffered 64x128 W-tiles
    __shared__ float lg[WG_TOKENS * LGSTRIDE];     // 64x64 logits (padded)

    const int tid  = threadIdx.x;
    const int lane = tid & 31;          // wave32
    const int wave = tid >> 5;          // 0..3
    const int tb   = blockIdx.x * WG_TOKENS;

    const int lrow  = lane & 15;        // M (A) / N (B) index within 16
    const int khalf = (lane >> 4) << 1; // K sub-offset 0 or 2 per ISA A/B layout

    v8f acc[4];
    #pragma unroll
    for (int t = 0; t < 4; ++t) acc[t] = (v8f){0.f,0.f,0.f,0.f,0.f,0.f,0.f,0.f};

#ifdef USE_TDM
    // ---------------- TDM double-buffered pipeline ----------------
    if (wave == 0) {  // TDM is per-wave (EXEC ignored): one issuing wave
        tdm_load_tile_2d(x + (size_t)tb * DIM, (unsigned)(uintptr_t)&xs[0][0],
                         DIM, 16384u, KC, WG_TOKENS, DIM);
        tdm_load_tile_2d(W,                    (unsigned)(uintptr_t)&ws[0][0],
                         DIM, NEXP,   KC, NEXP,      DIM);
    }
    for (int s = 0; s < NSTAGES; ++s) {
        const int cur = s & 1;
        if (wave == 0) {
            if (s + 1 < NSTAGES) {
                const int kb1 = (s + 1) * KC;
                tdm_load_tile_2d(x + (size_t)tb * DIM + kb1,
                                 (unsigned)(uintptr_t)&xs[cur ^ 1][0],
                                 DIM, 16384u, KC, WG_TOKENS, DIM);
                tdm_load_tile_2d(W + kb1,
                                 (unsigned)(uintptr_t)&ws[cur ^ 1][0],
                                 DIM, NEXP,   KC, NEXP,      DIM);
                // in-order TDM completion: <=2 outstanding => stage s finished
                __builtin_amdgcn_s_wait_tensorcnt(2);
            } else {
                __builtin_amdgcn_s_wait_tensorcnt(0);
            }
        }
        __syncthreads();   // stage s visible to all waves

        const float* xrow = &xs[cur][(wave * 16 + lrow) * XSTRIDE];
        const float* wsb  = &ws[cur][0];
        #pragma unroll
        for (int kk = 0; kk < KC; kk += 4) {
            v2f a  = *(const v2f*)&xrow[kk + khalf];
            v2f b0 = *(const v2f*)&wsb[(0 * 16 + lrow) * XSTRIDE + kk + khalf];
            v2f b1 = *(const v2f*)&wsb[(1 * 16 + lrow) * XSTRIDE + kk + khalf];
            v2f b2 = *(const v2f*)&wsb[(2 * 16 + lrow) * XSTRIDE + kk + khalf];
            v2f b3 = *(const v2f*)&wsb[(3 * 16 + lrow) * XSTRIDE + kk + khalf];
            acc[0] = __builtin_amdgcn_wmma_f32_16x16x4_f32(false, a, false, b0,
                                                           (short)0, acc[0], false, false);
            acc[1] = __builtin_amdgcn_wmma_f32_16x16x4_f32(false, a, false, b1,
                                                           (short)0, acc[1], false, false);
            acc[2] = __builtin_amdgcn_wmma_f32_16x16x4_f32(false, a, false, b2,
                                                           (short)0, acc[2], false, false);
            acc[3] = __builtin_amdgcn_wmma_f32_16x16x4_f32(false, a, false, b3,
                                                           (short)0, acc[3], false, false);
        }
        __syncthreads();   // all waves done with buf `cur` before it is re-filled
    }
#else
    // ---------------- fallback: synchronous staged pipeline ----------------
    for (int kb = 0; kb < DIM; kb += KC) {
        __syncthreads();
        #pragma unroll
        for (int i = 0; i < 16; ++i) {
            int j   = i * NTHREADS + tid;
            int row = j >> 5;
            int c4  = (j & 31) << 2;
            float4 vx = *(const float4*)&x[(size_t)(tb + row) * DIM + kb + c4];
            *(float4*)&xs[0][row * XSTRIDE + c4] = vx;
            float4 vw = *(const float4*)&W[(size_t)row * DIM + kb + c4];
            *(float4*)&ws[0][row * XSTRIDE + c4] = vw;
        }
        if (kb + KC < DIM) {
            int row = tid >> 1;
            int c   = (tid & 1) << 6;
            __builtin_prefetch(&x[(size_t)(tb + row) * DIM + kb + KC + c], 0, 3);
        }
        __syncthreads();

        const float* xrow = &xs[0][(wave * 16 + lrow) * XSTRIDE];
        #pragma unroll
        for (int kk = 0; kk < KC; kk += 4) {
            v2f a  = *(const v2f*)&xrow[kk + khalf];
            v2f b0 = *(const v2f*)&ws[0][(0 * 16 + lrow) * XSTRIDE + kk + khalf];
            v2f b1 = *(const v2f*)&ws[0][(1 * 16 + lrow) * XSTRIDE + kk + khalf];
            v2f b2 = *(const v2f*)&ws[0][(2 * 16 + lrow) * XSTRIDE + kk + khalf];
            v2f b3 = *(const v2f*)&ws[0][(3 * 16 + lrow) * XSTRIDE + kk + khalf];
            acc[0] = __builtin_amdgcn_wmma_f32_16x16x4_f32(false, a, false, b0,
                                                           (short)0, acc[0], false, false);
            acc[1] = __builtin_amdgcn_wmma_f32_16x16x4_f32(false, a, false, b1,
                                                           (short)0, acc[1], false, false);
            acc[2] = __builtin_amdgcn_wmma_f32_16x16x4_f32(false, a, false, b2,
                                                           (short)0, acc[2], false, false);
            acc[3] = __builtin_amdgcn_wmma_f32_16x16x4_f32(false, a, false, b3,
                                                           (short)0, acc[3], false, false);
        }
    }
#endif

    // ---- epilogue: bias add + spill logits to LDS per C/D VGPR layout ----
    // VGPR r, lanes 0-15: M=r, N=lane; lanes 16-31: M=r+8, N=lane-16
    const int hi = lane >> 4;
    #pragma unroll
    for (int t = 0; t < 4; ++t) {
        float bn = bias[t * 16 + lrow];
        #pragma unroll
        for (int r = 0; r < 8; ++r) {
            int m = r + 8 * hi;
            lg[(wave * 16 + m) * LGSTRIDE + (t * 16 + lrow)] = acc[t][r] + bn;
        }
    }
    __syncthreads();

    // ---- top-2 + softmax: one thread per token ----
    if (tid < WG_TOKENS) {
        const float* row = &lg[tid * LGSTRIDE];
        float v0 = -3.402823466e+38f, v1 = -3.402823466e+38f;
        int   i0 = 0, i1 = 0;
        #pragma unroll 8
        for (int e = 0; e < NEXP; ++e) {
            float v = row[e];
            if (v > v0)      { v1 = v0; i1 = i0; v0 = v; i0 = e; }
            else if (v > v1) { v1 = v;  i1 = e; }
        }
        float e1 = __expf(v1 - v0);   // v0 >= v1
        float d  = 1.0f + e1;
        size_t tok = (size_t)tb + (size_t)tid;
        out_probs[tok * 2 + 0] = 1.0f / d;
        out_probs[tok * 2 + 1] = e1 / d;
        out_idx[tok * 2 + 0] = i0;
        out_idx[tok * 2 + 1] = i1;
    }
}

extern "C" void kernel_launch(void* const* d_in, const int* in_sizes, int n_in,
                              void* d_out, int out_size, void* d_ws, size_t ws_size,
                              hipStream_t stream) {
    const float* x = (const float*)d_in[0];   // [4,4096,2048] f32
    const float* W = (const float*)d_in[1];   // [64,2048]     f32
    const float* b = (const float*)d_in[2];   // [64]          f32

    const int n_tokens = in_sizes[0] / DIM;   // 16384

    // d_out = [gate_probs: n_tokens*2 f32][topk_indices: n_tokens*2 i32]
    float* probs = (float*)d_out;
    int*   idx   = (int*)((float*)d_out + (size_t)n_tokens * 2);

    dim3 grid(n_tokens / WG_TOKENS);          // 256 blocks
    dim3 block(NTHREADS);                     // 4 waves
    moirai_gating_kernel<<<grid, block, 0, stream>>>(x, W, b, probs, idx);
}